// AttentiveDensenet_41927470744176
// MI455X (gfx1250) — compile-verified
//
#include <hip/hip_runtime.h>

typedef __attribute__((ext_vector_type(2))) float v2f;
typedef __attribute__((ext_vector_type(8))) float v8f;

__device__ __forceinline__ float lrelu(float v) { return v >= 0.0f ? v : 0.1f * v; }

// ---------------------------------------------------------------------------
// Generic 1x1-conv GEMM via V_WMMA_F32_16X16X4_F32.
//   Y[b,o,yo,xo] = sum_c W[o,c] * X[b,c,yo*sstr,xo*sstr] + bias[o]
// MODE 0: plain GEMM + bias.
// MODE 1: input transform t(v)=lrelu(v*scale[c]+shift[c]) (BN+LeakyReLU),
//         output epilogue Y = Xres + gamma*(GEMM + bias)   (residual write).
// All dims: O %16==0, C %4==0, N=B*Ho*Wo %16==0, Wo>=16 and %16==0.
// One 16x16 output tile per wave; 8 waves / 256-thread block.
// WMMA f32 16x16x4 lane layout assumed (mirrors ISA A-table / C-D striping):
//   A[m][k]: lane = m + 16*(k/2), vgpr = k%2
//   B[k][n]: lane = n + 16*(k/2), vgpr = k%2
//   D[m][n]: vgpr g -> m = g + 8*(lane/16), n = lane%16
// ---------------------------------------------------------------------------
template<int MODE>
__global__ __launch_bounds__(256)
void gemm1x1_wmma(const float* __restrict__ X, const float* __restrict__ W,
                  const float* __restrict__ bias, float* __restrict__ Y,
                  int C, int O, int Ho, int Wo, int sstr,
                  const float* __restrict__ ss,     // [0..127]=scale, [128..255]=shift (MODE 1)
                  const float* __restrict__ Xres,   // residual (MODE 1)
                  const float* __restrict__ gammap) // scalar gamma (MODE 1)
{
  const int HWo  = Ho * Wo;
  const int Win  = Wo * sstr;
  const int HWin = (Ho * sstr) * Win;
  const int tilesM = O >> 4;

  const int lane = threadIdx.x & 31;
  const int wave = threadIdx.x >> 5;
  const int ln = lane & 15;   // n (for B/D) and m (for A) sub-index
  const int kh = lane >> 4;   // K half selector

  const int tile = blockIdx.x * 8 + wave;
  const int mt = tile % tilesM;
  const int nt = tile / tilesM;

  const int n  = nt * 16 + ln;
  const int b  = n / HWo;
  const int p  = n - b * HWo;
  const int py = p / Wo;
  const int px = p - py * Wo;

  const float* __restrict__ xcol = X + (size_t)b * C * HWin
                                     + (size_t)(py * sstr) * Win + (size_t)(px * sstr);
  const float* __restrict__ wrow = W + (size_t)(mt * 16 + ln) * C;

  v8f acc = {0.f, 0.f, 0.f, 0.f, 0.f, 0.f, 0.f, 0.f};
  #pragma unroll 4
  for (int kb = 0; kb < C; kb += 4) {
    const int c0 = kb + 2 * kh;
    v2f a = *(const v2f*)(wrow + c0);                 // aligned: c0 even, C even
    float b0 = xcol[(size_t)c0 * HWin];
    float b1 = xcol[(size_t)(c0 + 1) * HWin];
    if (MODE == 1) {
      b0 = lrelu(b0 * ss[c0]     + ss[128 + c0]);
      b1 = lrelu(b1 * ss[c0 + 1] + ss[129 + c0]);
    }
    v2f bv; bv.x = b0; bv.y = b1;
    acc = __builtin_amdgcn_wmma_f32_16x16x4_f32(false, a, false, bv,
                                                (short)0, acc, false, false);
  }

  const float gamma = (MODE == 1) ? *gammap : 0.0f;
  #pragma unroll
  for (int g = 0; g < 8; ++g) {
    const int mm = mt * 16 + g + 8 * kh;
    const size_t oi = (size_t)b * O * HWo + (size_t)mm * HWo + (size_t)p;
    float v = acc[g] + bias[mm];
    if (MODE == 1) v = Xres[oi] + gamma * v;
    Y[oi] = v;
  }
}

// ---------------------------------------------------------------------------
// Per-pixel multi-source attention. One thread = one (batch, head, pixel).
// Sources s<nsrc are real; slot 4 is the implicit all-zero K/V entry.
// Unused slots get -1e30 logits (exp->0) so all loops are fixed-trip/unrolled.
// Top-4 threshold sparsification applies only when nsrc==4 (5 real entries).
// ---------------------------------------------------------------------------
__global__ __launch_bounds__(256)
void attn_kernel(const float* __restrict__ q,
                 const float* __restrict__ K0, const float* __restrict__ K1,
                 const float* __restrict__ K2, const float* __restrict__ K3,
                 const float* __restrict__ V0, const float* __restrict__ V1,
                 const float* __restrict__ V2, const float* __restrict__ V3,
                 int f0, int f1, int f2, int f3,
                 int nsrc, int h, int w, float* __restrict__ obuf)
{
  const int HW = h * w;
  const int tid = blockIdx.x * blockDim.x + threadIdx.x;  // ((b*8+head)*HW + p)
  const int p  = tid % HW;
  const int bh = tid / HW;
  const int head = bh & 7;
  const int b = bh >> 3;
  const int py = p / w;
  const int px = p - py * w;

  const float* Ks[4] = {K0, K1, K2, K3};
  const float* Vs[4] = {V0, V1, V2, V3};
  const int    fs[4] = {f0, f1, f2, f3};

  const size_t qbase = ((size_t)b * 128 + head * 16) * HW + p;
  float qv[16];
  #pragma unroll
  for (int c = 0; c < 16; ++c) qv[c] = q[qbase + (size_t)c * HW];

  float logits[5];
  #pragma unroll
  for (int s = 0; s < 4; ++s) {
    logits[s] = -1e30f;
    if (s < nsrc) {
      const int f  = fs[s];
      const int sw = w * f;
      const int sHW = (h * f) * sw;
      const size_t off = ((size_t)b * 128 + head * 16) * sHW
                       + (size_t)(py * f) * sw + (size_t)(px * f);
      const float* __restrict__ kp = Ks[s];
      float acc = 0.f;
      #pragma unroll
      for (int c = 0; c < 16; ++c) acc += qv[c] * kp[off + (size_t)c * sHW];
      logits[s] = acc * 0.25f;   // 1/TEMP, TEMP = sqrt(16)
    }
  }
  logits[4] = 0.0f;  // zero-entry (always present)

  float mx = logits[4];
  #pragma unroll
  for (int t = 0; t < 4; ++t) mx = fmaxf(mx, logits[t]);
  float attn[5];
  float sum = 0.f;
  #pragma unroll
  for (int t = 0; t < 5; ++t) { attn[t] = expf(logits[t] - mx); sum += attn[t]; }
  const float inv = 1.0f / sum;
  #pragma unroll
  for (int t = 0; t < 5; ++t) attn[t] *= inv;

  if (nsrc == 4) {  // 5 real entries: threshold at 4th-largest (= 2nd-smallest) + eps
    float m1 = 3.4e38f, m2 = 3.4e38f;
    #pragma unroll
    for (int t = 0; t < 5; ++t) {
      const float v = attn[t];
      if (v < m1) { m2 = m1; m1 = v; }
      else if (v < m2) { m2 = v; }
    }
    const float delta = m2 + 1e-7f;
    float s2 = 0.f;
    #pragma unroll
    for (int t = 0; t < 5; ++t) { attn[t] = fmaxf(attn[t] - delta, 0.f); s2 += attn[t]; }
    const float inv2 = 1.0f / (s2 + 1e-7f);
    #pragma unroll
    for (int t = 0; t < 5; ++t) attn[t] *= inv2;
  }

  float ov[16];
  #pragma unroll
  for (int vv = 0; vv < 16; ++vv) ov[vv] = 0.f;
  #pragma unroll
  for (int s = 0; s < 4; ++s) {
    if (s < nsrc) {
      const int f  = fs[s];
      const int sw = w * f;
      const int sHW = (h * f) * sw;
      const size_t off = ((size_t)b * 128 + head * 16) * sHW
                       + (size_t)(py * f) * sw + (size_t)(px * f);
      const float* __restrict__ vp = Vs[s];
      const float a = attn[s];
      #pragma unroll
      for (int vv = 0; vv < 16; ++vv) ov[vv] += a * vp[off + (size_t)vv * sHW];
    }
  }
  #pragma unroll
  for (int vv = 0; vv < 16; ++vv) obuf[qbase + (size_t)vv * HW] = ov[vv];
}

// ---------------------------------------------------------------------------
// BatchNorm statistics: one block per channel (C=128 fixed), biased variance,
// emits fused scale/shift:  scale=g*rsqrt(var+1e-3), shift=b-mean*scale.
// ---------------------------------------------------------------------------
__global__ __launch_bounds__(256)
void bn_stats(const float* __restrict__ o1, int Bn, int HW,
              const float* __restrict__ g, const float* __restrict__ bb,
              float* __restrict__ ss)
{
  const int c = blockIdx.x;                 // 0..127
  const int N = Bn * HW;
  float sum = 0.f, sq = 0.f;
  for (int i = threadIdx.x; i < N; i += 256) {
    const int b = i / HW;
    const int p = i - b * HW;
    const float v = o1[((size_t)b * 128 + c) * HW + p];
    sum += v; sq += v * v;
  }
  __shared__ float s1[256], s2[256];
  s1[threadIdx.x] = sum; s2[threadIdx.x] = sq;
  __syncthreads();
  for (int st = 128; st > 0; st >>= 1) {
    if (threadIdx.x < st) {
      s1[threadIdx.x] += s1[threadIdx.x + st];
      s2[threadIdx.x] += s2[threadIdx.x + st];
    }
    __syncthreads();
  }
  if (threadIdx.x == 0) {
    const float mean = s1[0] / (float)N;
    const float var  = s2[0] / (float)N - mean * mean;  // biased, like torch BN train
    const float sc = g[c] * rsqrtf(var + 1e-3f);
    ss[c] = sc;
    ss[128 + c] = bb[c] - mean * sc;
  }
}

// ---------------------------------------------------------------------------
extern "C" void kernel_launch(void* const* d_in, const int* in_sizes, int n_in,
                              void* d_out, int out_size, void* d_ws, size_t ws_size,
                              hipStream_t stream) {
  (void)in_sizes; (void)n_in; (void)out_size; (void)ws_size;

  const float* x[4];
  for (int i = 0; i < 4; ++i) x[i] = (const float*)d_in[i];
  // params flattened by sorted key: Wk, Wo1, Wo2, Wq, Wv, bk, bn_b, bn_g, bo1, bo2, bq, bv, gamma
  const float *Wk[4], *Wo1[4], *Wo2[4], *Wq[4], *Wv[4];
  const float *bk[4], *bnb[4], *bng[4], *bo1[4], *bo2[4], *bq[4], *bv[4], *gam[4];
  for (int i = 0; i < 4; ++i) {
    Wk[i]  = (const float*)d_in[ 4 + i];
    Wo1[i] = (const float*)d_in[ 8 + i];
    Wo2[i] = (const float*)d_in[12 + i];
    Wq[i]  = (const float*)d_in[16 + i];
    Wv[i]  = (const float*)d_in[20 + i];
    bk[i]  = (const float*)d_in[24 + i];
    bnb[i] = (const float*)d_in[28 + i];
    bng[i] = (const float*)d_in[32 + i];
    bo1[i] = (const float*)d_in[36 + i];
    bo2[i] = (const float*)d_in[40 + i];
    bq[i]  = (const float*)d_in[44 + i];
    bv[i]  = (const float*)d_in[48 + i];
    gam[i] = (const float*)d_in[52 + i];
  }

  float* ws = (float*)d_ws;
  const size_t SZ64 = (size_t)16 * 128 * 64 * 64;   // 8,388,608 floats
  const size_t SZ32 = (size_t)16 * 128 * 32 * 32;   // 2,097,152 floats
  float *kb_[4], *vb_[4];
  kb_[0] = ws;                vb_[0] = ws + SZ64;
  kb_[1] = ws + 2 * SZ64;     vb_[1] = ws + 3 * SZ64;
  kb_[2] = ws + 4 * SZ64;     vb_[2] = kb_[2] + SZ32;
  kb_[3] = vb_[2] + SZ32;     vb_[3] = kb_[3] + SZ32;
  float* qb  = vb_[3] + SZ32;
  float* ob  = qb  + SZ64;
  float* o1b = ob  + SZ64;
  float* ssb = o1b + SZ64;    // 256 floats

  const int CH[4]  = {64, 128, 256, 256};
  const int HH[4]  = {128, 64, 32, 32};
  const int KVH[4] = {64, 64, 32, 32};   // k/v stored grid (layer0 pre-subsampled 2x)

  // ---- write phase: k_i, v_i at stored resolution (conv commutes with sampling)
  for (int i = 0; i < 4; ++i) {
    const int Hs = KVH[i];
    const int sstr = HH[i] / Hs;
    const int tiles = (128 / 16) * ((16 * Hs * Hs) / 16);
    dim3 grid(tiles / 8), blk(256);
    gemm1x1_wmma<0><<<grid, blk, 0, stream>>>(x[i], Wk[i], bk[i], kb_[i],
        CH[i], 128, Hs, Hs, sstr, nullptr, nullptr, nullptr);
    gemm1x1_wmma<0><<<grid, blk, 0, stream>>>(x[i], Wv[i], bv[i], vb_[i],
        CH[i], 128, Hs, Hs, sstr, nullptr, nullptr, nullptr);
  }

  // ---- read phase: layers 1..3
  size_t outoff = 0;
  for (int r = 1; r < 4; ++r) {
    const int h  = HH[r];
    const int HW = h * h;

    {  // q projection
      const int tiles = 8 * ((16 * HW) / 16);
      gemm1x1_wmma<0><<<dim3(tiles / 8), 256, 0, stream>>>(x[r], Wq[r], bq[r], qb,
          CH[r], 128, h, h, 1, nullptr, nullptr, nullptr);
    }
    {  // attention read over sources 0..r (+ implicit zero entry)
      int f[4] = {1, 1, 1, 1};
      for (int s = 0; s <= r; ++s) f[s] = KVH[s] / h;   // nearest-resize factor
      const int nthreads = 16 * 8 * HW;
      attn_kernel<<<dim3(nthreads / 256), 256, 0, stream>>>(qb,
          kb_[0], kb_[1], kb_[2], kb_[3],
          vb_[0], vb_[1], vb_[2], vb_[3],
          f[0], f[1], f[2], f[3], r + 1, h, h, ob);
    }
    {  // Wo1 projection
      const int tiles = 8 * ((16 * HW) / 16);
      gemm1x1_wmma<0><<<dim3(tiles / 8), 256, 0, stream>>>(ob, Wo1[r], bo1[r], o1b,
          128, 128, h, h, 1, nullptr, nullptr, nullptr);
    }
    // BN stats -> fused scale/shift
    bn_stats<<<dim3(128), 256, 0, stream>>>(o1b, 16, HW, bng[r], bnb[r], ssb);
    {  // Wo2 with BN+LeakyReLU on input side, residual + gamma on output side
      const int O = CH[r];
      const int tiles = (O / 16) * ((16 * HW) / 16);
      gemm1x1_wmma<1><<<dim3(tiles / 8), 256, 0, stream>>>(o1b, Wo2[r], bo2[r],
          (float*)d_out + outoff, 128, O, h, h, 1, ssb, x[r], gam[r]);
    }
    outoff += (size_t)16 * CH[r] * HW;
  }
}